// TemporalMemory_82884278878367
// MI455X (gfx1250) — compile-verified
//
#include <hip/hip_runtime.h>

// ---------------- problem constants (match reference) ----------------
#define NCELLS   16384      // N = COLUMNS * CPC
#define NCOLS    1024
#define CPC      16
#define NSEG     32
#define NSYN     128
#define ACT_TH   10
#define CONN_PERM 0.8f
#define NBATCH   4

#define GLOBAL_AS __attribute__((address_space(1)))
#define LDS_AS    __attribute__((address_space(3)))

typedef int   v4i __attribute__((ext_vector_type(4)));
typedef float v4f __attribute__((ext_vector_type(4)));

typedef GLOBAL_AS v4i* gv4i_p;   // AS1 pointer-to-v4i (async builtin param 1)
typedef LDS_AS    v4i* lv4i_p;   // AS3 pointer-to-v4i (async builtin param 2)

// Stage the 16KB packed-activity table into LDS via the CDNA5 async path
// (GLOBAL_LOAD_ASYNC_TO_LDS_B128, tracked by ASYNCcnt).
__device__ __forceinline__ void stage_activity(const unsigned char* packedActive,
                                               unsigned char* act /* LDS, 16KB */) {
    const unsigned t = threadIdx.x;
#pragma unroll
    for (int i = 0; i < 4; ++i) {
        const unsigned off = (t + i * 256u) * 16u;
        __builtin_amdgcn_global_load_async_to_lds_b128(
            (gv4i_p)(packedActive + off),
            (lv4i_p)(act + off), 0, 0);
    }
    __builtin_amdgcn_s_wait_asynccnt(0);
    __syncthreads();
}

// Byte-wise wave32 reduction + threshold -> 4 predictive bits.
__device__ __forceinline__ unsigned reduce_threshold(unsigned acc) {
#pragma unroll
    for (int o = 16; o > 0; o >>= 1) acc += __shfl_xor(acc, o, 32);
    unsigned ge = 0;
    ge |= ((acc        & 0xFFu) >= ACT_TH) ? 1u : 0u;
    ge |= (((acc >> 8) & 0xFFu) >= ACT_TH) ? 2u : 0u;
    ge |= (((acc >> 16)& 0xFFu) >= ACT_TH) ? 4u : 0u;
    ge |= (((acc >> 24)       ) >= ACT_TH) ? 8u : 0u;
    return ge;
}

// ---------------------------------------------------------------------
// Kernel 1: pack prev_active (4 x N bool bytes) into one byte per cell
// (bit b = batch b active). Also zero the accuracy counters.
// ---------------------------------------------------------------------
__global__ __launch_bounds__(256) void tm_pack_prev(
    const unsigned char* __restrict__ prev,   // [4][NCELLS] bool
    unsigned char* __restrict__ packed,       // [NCELLS]
    int* __restrict__ counters)               // [8]
{
    const int n = blockIdx.x * 256 + threadIdx.x;   // grid is exactly NCELLS
    unsigned b = 0;
#pragma unroll
    for (int bt = 0; bt < NBATCH; ++bt)
        b |= (unsigned)(prev[bt * NCELLS + n] != 0) << bt;
    packed[n] = (unsigned char)b;
    if (n < 8) counters[n] = 0;
}

// ---------------------------------------------------------------------
// Kernel 2a: PHASE 1 segments. One wave per cell, lane owns 4 synapses
// of the current segment (coalesced b128 NT streams of conn/vol/cons).
// Computes packed 4-batch overlaps AND materializes the connected mask
// (1 byte per (segment,lane) = 4 bits used) so phase 2 never touches
// the 512MB of permanences again.
// ---------------------------------------------------------------------
__global__ __launch_bounds__(256) void tm_segments_p1(
    const int*   __restrict__ conn,           // [NCELLS][NSEG][NSYN]
    const float* __restrict__ vol,
    const float* __restrict__ cons,
    const unsigned char* __restrict__ packedActive,  // [NCELLS]
    unsigned char* __restrict__ predOut,      // [NCELLS] nibble per cell
    unsigned char* __restrict__ cmaskOut)     // [NCELLS*NSEG*32]
{
    __shared__ __align__(16) unsigned char act[NCELLS];   // 16 KB
    stage_activity(packedActive, act);

    const int lane = threadIdx.x & 31;
    const int cell = (blockIdx.x * 256 + threadIdx.x) >> 5;

    unsigned predNib = 0;
    for (int s = 0; s < NSEG; ++s) {
        const size_t seg  = (size_t)cell * NSEG + s;
        const size_t base = seg * NSYN + (size_t)lane * 4;
        v4i c = __builtin_nontemporal_load((const v4i*)(conn + base));
        v4f v = __builtin_nontemporal_load((const v4f*)(vol  + base));
        v4f q = __builtin_nontemporal_load((const v4f*)(cons + base));

        unsigned acc = 0, cmask = 0;
#pragma unroll
        for (int j = 0; j < 4; ++j) {
            const bool con = (v[j] + q[j]) >= CONN_PERM;
            cmask |= (unsigned)con << j;
            const unsigned nib = act[c[j] & (NCELLS - 1)];
            const unsigned m = con ? nib : 0u;
            acc += (m * 0x204081u) & 0x01010101u;   // spread b0..b3 into bytes
        }
        cmaskOut[seg * 32 + lane] = (unsigned char)cmask;  // default TH: keep in L2
        predNib |= reduce_threshold(acc);
        // no early break: every segment's cmask must be written for phase 2
    }
    if (lane == 0) predOut[cell] = (unsigned char)predNib;
}

// ---------------------------------------------------------------------
// Kernel 2b: PHASE 2 segments. Reads conn + cached connected mask only
// (272MB instead of 768MB; cmask is L2-resident).
// ---------------------------------------------------------------------
__global__ __launch_bounds__(256) void tm_segments_p2(
    const int*   __restrict__ conn,
    const unsigned char* __restrict__ cmaskIn,       // [NCELLS*NSEG*32]
    const unsigned char* __restrict__ packedActive,  // [NCELLS]
    unsigned char* __restrict__ predOut)             // [NCELLS]
{
    __shared__ __align__(16) unsigned char act[NCELLS];   // 16 KB
    stage_activity(packedActive, act);

    const int lane = threadIdx.x & 31;
    const int cell = (blockIdx.x * 256 + threadIdx.x) >> 5;

    unsigned predNib = 0;
    for (int s = 0; s < NSEG; ++s) {
        const size_t seg  = (size_t)cell * NSEG + s;
        const size_t base = seg * NSYN + (size_t)lane * 4;
        v4i c = __builtin_nontemporal_load((const v4i*)(conn + base));
        const unsigned cmask = cmaskIn[seg * 32 + lane];

        unsigned acc = 0;
#pragma unroll
        for (int j = 0; j < 4; ++j) {
            const unsigned nib = act[c[j] & (NCELLS - 1)];
            const unsigned m = ((cmask >> j) & 1u) ? nib : 0u;
            acc += (m * 0x204081u) & 0x01010101u;
        }
        predNib |= reduce_threshold(acc);
        if (predNib == 0xFu) break;   // all 4 batches predictive: any() satisfied
    }
    if (lane == 0) predOut[cell] = (unsigned char)predNib;
}

// ---------------------------------------------------------------------
// Kernel 2c: fallback fused variant (used for both phases if ws_size is
// too small for the 16MB connected-mask cache).
// ---------------------------------------------------------------------
__global__ __launch_bounds__(256) void tm_segments_full(
    const int*   __restrict__ conn,
    const float* __restrict__ vol,
    const float* __restrict__ cons,
    const unsigned char* __restrict__ packedActive,
    unsigned char* __restrict__ predOut)
{
    __shared__ __align__(16) unsigned char act[NCELLS];
    stage_activity(packedActive, act);

    const int lane = threadIdx.x & 31;
    const int cell = (blockIdx.x * 256 + threadIdx.x) >> 5;

    unsigned predNib = 0;
    for (int s = 0; s < NSEG; ++s) {
        const size_t base = ((size_t)cell * NSEG + s) * NSYN + (size_t)lane * 4;
        v4i c = __builtin_nontemporal_load((const v4i*)(conn + base));
        v4f v = __builtin_nontemporal_load((const v4f*)(vol  + base));
        v4f q = __builtin_nontemporal_load((const v4f*)(cons + base));

        unsigned acc = 0;
#pragma unroll
        for (int j = 0; j < 4; ++j) {
            const unsigned nib = act[c[j] & (NCELLS - 1)];
            const unsigned m = ((v[j] + q[j]) >= CONN_PERM) ? nib : 0u;
            acc += (m * 0x204081u) & 0x01010101u;
        }
        predNib |= reduce_threshold(acc);
        if (predNib == 0xFu) break;
    }
    if (lane == 0) predOut[cell] = (unsigned char)predNib;
}

// ---------------------------------------------------------------------
// Kernel 3: per-column phase-1 combine: winner/burst, new_active output
// (floats), packed new-active bytes for phase 2, accuracy counters.
// ---------------------------------------------------------------------
__global__ __launch_bounds__(256) void tm_combine(
    const unsigned char* __restrict__ predCell,   // [NCELLS] nibbles
    const unsigned char* __restrict__ sdr,        // [4][NCOLS] bool
    float* __restrict__ newActiveOut,             // [4][NCELLS]
    unsigned char* __restrict__ packedB,          // [NCELLS]
    int* __restrict__ counters)                   // [0..3]=num_pred [4..7]=num_act
{
    const int col = blockIdx.x * 256 + threadIdx.x;
    if (col >= NCOLS) return;

    unsigned nib[CPC];
#pragma unroll
    for (int i = 0; i < CPC; ++i) nib[i] = predCell[col * CPC + i];

    unsigned char pb[CPC];
#pragma unroll
    for (int i = 0; i < CPC; ++i) pb[i] = 0;

#pragma unroll
    for (int b = 0; b < NBATCH; ++b) {
        const unsigned sd = (sdr[b * NCOLS + col] != 0) ? 1u : 0u;
        unsigned colPred = 0;
#pragma unroll
        for (int i = 0; i < CPC; ++i) colPred |= (nib[i] >> b) & 1u;
#pragma unroll
        for (int i = 0; i < CPC; ++i) {
            const unsigned w = colPred ? ((nib[i] >> b) & 1u) : 1u;  // burst if no pred
            const unsigned a = w & sd;                               // only active cols
            newActiveOut[(size_t)b * NCELLS + col * CPC + i] = (float)a;
            pb[i] |= (unsigned char)(a << b);
        }
        if (sd) {
            atomicAdd(&counters[b], (int)colPred);
            atomicAdd(&counters[4 + b], 1);
        }
    }
#pragma unroll
    for (int i = 0; i < CPC; ++i) packedB[col * CPC + i] = pb[i];
}

// ---------------------------------------------------------------------
// Kernel 4: finalize: pred_next = phase-2 predictive & sdr; accuracy.
// ---------------------------------------------------------------------
__global__ __launch_bounds__(256) void tm_finalize(
    const unsigned char* __restrict__ predCell2,  // [NCELLS] nibbles
    const unsigned char* __restrict__ sdr,        // [4][NCOLS] bool
    float* __restrict__ predNextOut,              // [4][NCELLS]
    float* __restrict__ accOut,                   // [4]
    const int* __restrict__ counters)
{
    const int col = blockIdx.x * 256 + threadIdx.x;
    if (col < NCOLS) {
#pragma unroll
        for (int b = 0; b < NBATCH; ++b) {
            const unsigned sd = (sdr[b * NCOLS + col] != 0) ? 1u : 0u;
#pragma unroll
            for (int i = 0; i < CPC; ++i) {
                const unsigned p = (predCell2[col * CPC + i] >> b) & 1u;
                predNextOut[(size_t)b * NCELLS + col * CPC + i] = (float)(p & sd);
            }
        }
    }
    if (blockIdx.x == 0 && threadIdx.x < NBATCH) {
        const int np = counters[threadIdx.x];
        const int na = counters[4 + threadIdx.x];
        accOut[threadIdx.x] = (na > 0) ? ((float)np / (float)(na > 1 ? na : 1)) : 1.0f;
    }
}

// ---------------------------------------------------------------------
extern "C" void kernel_launch(void* const* d_in, const int* in_sizes, int n_in,
                              void* d_out, int out_size, void* d_ws, size_t ws_size,
                              hipStream_t stream) {
    // setup_inputs() order:
    // 0 sdr_batch [4,1024] bool      1 prev_active_cells [4,16384] bool
    // 2 distal_connections i32       3 volatile_permanences f32
    // 4 consolidated_permanences f32 5 modulation_signal_batch f32 (unused)
    const unsigned char* sdr  = (const unsigned char*)d_in[0];
    const unsigned char* prev = (const unsigned char*)d_in[1];
    const int*   conn = (const int*)d_in[2];
    const float* vol  = (const float*)d_in[3];
    const float* cons = (const float*)d_in[4];

    unsigned char* ws = (unsigned char*)d_ws;
    unsigned char* pA    = ws;                 // packed prev activity   [16K]
    unsigned char* predA = ws + 1 * NCELLS;    // phase-1 nibbles        [16K]
    unsigned char* pB    = ws + 2 * NCELLS;    // packed new activity    [16K]
    unsigned char* predB = ws + 3 * NCELLS;    // phase-2 nibbles        [16K]
    int* counters        = (int*)(ws + 4 * NCELLS);   // 8 ints @ 64K
    unsigned char* cmask = ws + 4 * NCELLS + 64;      // 16MB connected-mask cache

    const size_t needWs = (size_t)4 * NCELLS + 64 + (size_t)NCELLS * NSEG * 32;
    const bool useCache = ws_size >= needWs;

    float* outNewActive = (float*)d_out;                         // [4*16384]
    float* outPredNext  = outNewActive + NBATCH * NCELLS;        // [4*16384]
    float* outAcc       = outPredNext  + NBATCH * NCELLS;        // [4]

    tm_pack_prev<<<NCELLS / 256, 256, 0, stream>>>(prev, pA, counters);

    if (useCache) {
        tm_segments_p1<<<(NCELLS * 32) / 256, 256, 0, stream>>>(conn, vol, cons, pA, predA, cmask);
    } else {
        tm_segments_full<<<(NCELLS * 32) / 256, 256, 0, stream>>>(conn, vol, cons, pA, predA);
    }

    tm_combine<<<NCOLS / 256, 256, 0, stream>>>(predA, sdr, outNewActive, pB, counters);

    if (useCache) {
        tm_segments_p2<<<(NCELLS * 32) / 256, 256, 0, stream>>>(conn, cmask, pB, predB);
    } else {
        tm_segments_full<<<(NCELLS * 32) / 256, 256, 0, stream>>>(conn, vol, cons, pB, predB);
    }

    tm_finalize<<<NCOLS / 256, 256, 0, stream>>>(predB, sdr, outPredNext, outAcc, counters);
}